// LA_283467842629
// MI455X (gfx1250) — compile-verified
//
#include <hip/hip_runtime.h>

#define C_TOT   2048
#define H_HEADS 32
#define CH      64
#define SP      36                  // 6x6
#define CHW     (C_TOT * SP)        // 73728 floats = 288 KB per batch element
#define HEADF   (CH * SP)           // 2304 floats per head region (9 KB)
#define THREADS 1024
#define WCHUNKS (HEADF / 4 / 32)    // 18 b128 chunks per lane (per-wave head copy)

// ---- CDNA5 async global<->LDS (ASYNCcnt path), signature probe-verified ----
typedef int v4i __attribute__((ext_vector_type(4)));
typedef __attribute__((address_space(1))) v4i* gv4i_ptr;
typedef __attribute__((address_space(3))) v4i* lv4i_ptr;

__device__ __forceinline__ void async_g2l_b128(const float* g, float* l) {
#if __has_builtin(__builtin_amdgcn_global_load_async_to_lds_b128)
    float* gm = const_cast<float*>(g);
    __builtin_amdgcn_global_load_async_to_lds_b128((gv4i_ptr)gm, (lv4i_ptr)l, 0, 0);
#else
    *(float4*)l = *(const float4*)g;
#endif
}

__device__ __forceinline__ void async_l2g_b128(float* g, float* l) {
#if __has_builtin(__builtin_amdgcn_global_store_async_from_lds_b128)
    __builtin_amdgcn_global_store_async_from_lds_b128((gv4i_ptr)g, (lv4i_ptr)l, 0, 0);
#else
    *(float4*)g = *(const float4*)l;
#endif
}

__device__ __forceinline__ void wait_async0() {
#if __has_builtin(__builtin_amdgcn_s_wait_asynccnt)
    __builtin_amdgcn_s_wait_asynccnt(0);
#else
    asm volatile("s_wait_asynccnt 0" ::: "memory");
#endif
}

// Order own-wave ds_store -> async store-from-LDS (replaces a block barrier).
__device__ __forceinline__ void wait_ds0() {
    asm volatile("" ::: "memory");
#if __has_builtin(__builtin_amdgcn_s_wait_dscnt)
    __builtin_amdgcn_s_wait_dscnt(0);
#else
    asm volatile("s_wait_dscnt 0" ::: "memory");
#endif
}

// One workgroup per batch element; wave w == head h (wave32); lane owns channels 2l, 2l+1.
// Wave h stages/reduces/gates/stores ONLY its own head region -> per-wave async
// counters give the ordering; block barriers needed only around the cross-head squeeze.
__global__ __launch_bounds__(THREADS)
void se_gate_fused_kernel(const float* __restrict__ x,
                          const float* __restrict__ f1_w, const float* __restrict__ f1_b,
                          const float* __restrict__ f2_w, const float* __restrict__ f2_b,
                          const float* __restrict__ v_w,  const float* __restrict__ v_b,
                          float* __restrict__ out) {
    extern __shared__ float smem[];
    float* xbuf  = smem;                    // 73728 floats (288 KB)
    float* smean = xbuf + CHW;              // 32*9
    float* smax  = smean + H_HEADS * 9;     // 32*9
    float* sse   = smax + H_HEADS * 9;      // 18 ([0..8]=mean path, [9..17]=max path)

    const int tid = threadIdx.x;
    const int h   = tid >> 5;               // wave index == head
    const int l   = tid & 31;

    const float* xg   = x   + (size_t)blockIdx.x * CHW + h * HEADF;
    float*       og   = out + (size_t)blockIdx.x * CHW + h * HEADF;
    float*       hbuf = xbuf + h * HEADF;

    // ---- Phase A: wave h async-loads its own head region (512 B / instruction) ----
#pragma unroll
    for (int j = 0; j < WCHUNKS; ++j) {
        const int off = (j * 32 + l) * 4;
        async_g2l_b128(xg + off, hbuf + off);
    }
    wait_async0();   // per-wave counter: this head's 9 KB is now in LDS

    // ---- Phase B: per-head 3x3 mean/max over (64 ch x 2 x 2), B128 LDS reads ----
    float s[9], m[9];
#pragma unroll
    for (int k = 0; k < 9; ++k) { s[k] = 0.f; m[k] = -__builtin_inff(); }
#pragma unroll
    for (int c2 = 0; c2 < 2; ++c2) {
        const float* p = hbuf + (2 * l + c2) * SP;   // 144 B, 16-B aligned
#pragma unroll
        for (int q = 0; q < 9; ++q) {
            const float4 v = *(const float4*)(p + 4 * q);
            const float e[4] = {v.x, v.y, v.z, v.w};
#pragma unroll
            for (int t = 0; t < 4; ++t) {
                const int pos = 4 * q + t;                       // row-major 6x6
                const int k   = ((pos / 6) >> 1) * 3 + ((pos % 6) >> 1);
                s[k] += e[t];
                m[k] = fmaxf(m[k], e[t]);
            }
        }
    }
    // wave32 shuffle tree: 32 lanes -> full 256-element stats
#pragma unroll
    for (int d = 16; d >= 1; d >>= 1) {
#pragma unroll
        for (int k = 0; k < 9; ++k) {
            s[k] += __shfl_xor(s[k], d, 32);
            m[k]  = fmaxf(m[k], __shfl_xor(m[k], d, 32));
        }
    }
    if (l == 0) {
#pragma unroll
        for (int k = 0; k < 9; ++k) {
            smean[h * 9 + k] = s[k] * (1.f / 256.f);
            smax [h * 9 + k] = m[k];
        }
    }
    __syncthreads();

    // ---- Phase C: squeeze (1x1 conv H->1) + ReLU for mean & max paths ----
    if (tid < 18) {
        const int k = tid % 9;
        const float* src = (tid < 9) ? smean : smax;
        float acc = f1_b[0];
#pragma unroll 8
        for (int hh = 0; hh < H_HEADS; ++hh) acc += src[hh * 9 + k] * f1_w[hh];
        sse[tid] = fmaxf(acc, 0.f);
    }
    __syncthreads();

    // ---- Phase D: excite + fuse + bilinear 3x3->6x6 + sigmoid, apply in LDS ----
    const float v0 = v_w[0], v1 = v_w[1], vb = v_b[0];
    const float w2 = f2_w[h], b2 = f2_b[h];
    float fus[9];
#pragma unroll
    for (int k = 0; k < 9; ++k)
        fus[k] = (sse[k] * v0 + sse[9 + k] * v1) * w2 + b2 * (v0 + v1) + vb;

    // PyTorch bilinear (align_corners=False), 3->6: src=max(0,(d+0.5)*0.5-0.5)
    const int   i0[6] = {0, 0, 0, 1, 1, 2};
    const int   i1[6] = {0, 1, 1, 2, 2, 2};
    const float wf[6] = {0.f, 0.25f, 0.75f, 0.25f, 0.75f, 0.25f};

    float rowv[6][3];
#pragma unroll
    for (int r = 0; r < 6; ++r)
#pragma unroll
        for (int c = 0; c < 3; ++c)
            rowv[r][c] = fus[i0[r] * 3 + c] * (1.f - wf[r]) + fus[i1[r] * 3 + c] * wf[r];

    float scl[36];
#pragma unroll
    for (int r = 0; r < 6; ++r)
#pragma unroll
        for (int cc = 0; cc < 6; ++cc) {
            const float g = rowv[r][i0[cc]] * (1.f - wf[cc]) + rowv[r][i1[cc]] * wf[cc];
            scl[r * 6 + cc] = 1.f + 1.f / (1.f + __expf(-g));   // 1 + sigmoid
        }

#pragma unroll
    for (int c2 = 0; c2 < 2; ++c2) {
        float* p = hbuf + (2 * l + c2) * SP;
#pragma unroll
        for (int q = 0; q < 9; ++q) {
            float4 v = *(float4*)(p + 4 * q);
            v.x *= scl[4 * q + 0];
            v.y *= scl[4 * q + 1];
            v.z *= scl[4 * q + 2];
            v.w *= scl[4 * q + 3];
            *(float4*)(p + 4 * q) = v;
        }
    }
    wait_ds0();      // own-wave LDS writes committed before async engine reads them

    // ---- Phase E: wave h async-stores its own head region (coalesced B128) ----
#pragma unroll
    for (int j = 0; j < WCHUNKS; ++j) {
        const int off = (j * 32 + l) * 4;
        async_l2g_b128(og + off, hbuf + off);
    }
    wait_async0();
}

extern "C" void kernel_launch(void* const* d_in, const int* in_sizes, int n_in,
                              void* d_out, int out_size, void* d_ws, size_t ws_size,
                              hipStream_t stream) {
    const float* x   = (const float*)d_in[0];
    const float* f1w = (const float*)d_in[1];
    const float* f1b = (const float*)d_in[2];
    const float* f2w = (const float*)d_in[3];
    const float* f2b = (const float*)d_in[4];
    const float* vw  = (const float*)d_in[5];
    const float* vb  = (const float*)d_in[6];
    float* out = (float*)d_out;

    const int B = in_sizes[0] / CHW;                       // 512
    const size_t shmem = (size_t)(CHW + 2 * H_HEADS * 9 + 18) * sizeof(float); // ~291 KB

    se_gate_fused_kernel<<<B, THREADS, shmem, stream>>>(x, f1w, f1b, f2w, f2b, vw, vb, out);

    (void)n_in; (void)out_size; (void)d_ws; (void)ws_size;
}